// SGNS_70317204570605
// MI455X (gfx1250) — compile-verified
//
#include <hip/hip_runtime.h>

// ---------------------------------------------------------------------------
// SGNS loss for MI455X (gfx1250, wave32).
// Memory-bound gather workload (~0.5 FLOP/byte): ~440 MB of embedding-row
// gathers vs 0.22 GFLOP of dots; HBM floor ~19 us (less if L2-resident:
// both tables = 102 MB < 192 MB L2). Strategy:
//  * one wave32 per batch item; all 14 tile word-indices gathered in ONE
//    clause up front (kills the per-tile serialized index-load stall seen
//    in the previous build), row pointers precomputed.
//  * V_WMMA_F32_16X16X4_F32 computes 16 dots/tile with zero cross-lane
//    shuffles: A = 16 context rows (ISA A layout: lane m%16 holds row m,
//    K halves split across lane halves), B = iv[b] replicated in all 16
//    columns. Only the diagonal of D is consumed — wasted FLOPs are free
//    at this arithmetic intensity. Dual accumulators for pipeline depth.
//  * tile loop fully unrolled so the scheduler overlaps tile t+1 loads with
//    tile t WMMAs; rolling global_prefetch_b8 one tile ahead (2 prefetches
//    per lane cover all four 128B lines of a 512B row).
//  * deterministic reduction: wave shfl -> LDS -> per-block partial in d_ws
//    -> single-block tree reduce -> d_out[0] = -(sum)/(C*B)
// ---------------------------------------------------------------------------

typedef float v2f __attribute__((ext_vector_type(2)));
typedef float v8f __attribute__((ext_vector_type(8)));

#define B_SZ   4096
#define DIM    128
#define C_CTX  10
#define NNEG   200            // C * N_NEGS
#define NCTX   210            // C + C*N_NEGS
#define NTILES 14             // ceil(210/16)
#define WPB    8              // waves per block
#define NBLK   (B_SZ / WPB)   // 512

__device__ __forceinline__ float log_sigmoid(float x) {
    // stable: min(x,0) - log1p(exp(-|x|))
    float ax = __builtin_fabsf(x);
    float e  = __expf(-ax);
    return fminf(x, 0.0f) - __logf(1.0f + e);
}

__global__ __launch_bounds__(256) void sgns_score_kernel(
    const float* __restrict__ emb_i, const float* __restrict__ emb_o,
    const int* __restrict__ iword, const int* __restrict__ owords,
    const int* __restrict__ nwords, float* __restrict__ partial)
{
    __shared__ float wave_sum[WPB];

    const int lane = threadIdx.x & 31;
    const int wave = threadIdx.x >> 5;
    const int b    = blockIdx.x * WPB + wave;

    const int half = lane >> 4;   // K-half select (lanes 0-15 vs 16-31)
    const int mrow = lane & 15;   // A-matrix row index within a tile

    // ---- B operand: iv[b] broadcast into all 16 columns.
    // Lane l's two B dwords for K-step s are the contiguous pair
    // iv[4s + 2*half .. +1] (same K mapping as the A operand).
    const float* ivRow = emb_i + (size_t)iword[b] * DIM;
    v2f bfrag[32];
#pragma unroll
    for (int s = 0; s < 32; ++s)
        bfrag[s] = *(const v2f*)(ivRow + 4 * s + 2 * half);

    // ---- Preload ALL tile word indices in one clause (single gather
    // latency instead of one per tile) and precompute row pointers.
    const float* rowp[NTILES];
#pragma unroll
    for (int t = 0; t < NTILES; ++t) {
        int j  = t * 16 + mrow;                 // context index 0..223
        int jc = (j < NCTX - 1) ? j : (NCTX - 1);
        int widx = (jc < C_CTX) ? owords[b * C_CTX + jc]
                                : nwords[b * NNEG + (jc - C_CTX)];
        rowp[t] = emb_o + (size_t)widx * DIM;
    }

    // Warm tile 0 rows (lanes l and l+16 share a row: each lane covers two
    // of its row's four 128B lines).
    {
        const char* p = (const char*)rowp[0] + half * 256;
        __builtin_prefetch(p, 0, 3);
        __builtin_prefetch(p + 128, 0, 3);
    }

    // diag ownership: lanes 0-7 hold D[m][m], m=0..7, at acc[r=lane];
    // lanes 24-31 hold m=8..15 at acc[r=lane-24].
    const bool active = ((((lane >> 3) ^ (lane >> 4)) & 1) == 0);
    const int  m      = (lane & 7) + ((lane >> 4) << 3);
    const int  r      = lane & 7;

    float lsum = 0.0f;

#pragma unroll
    for (int t = 0; t < NTILES; ++t) {
        // Rolling prefetch one tile ahead (address known up front -> no
        // dependent-load stall before issuing).
        if (t + 1 < NTILES) {
            const char* p = (const char*)rowp[t + 1] + half * 256;
            __builtin_prefetch(p, 0, 3);
            __builtin_prefetch(p + 128, 0, 3);
        }

        const float* ctxRow = rowp[t];
        v8f acc0 = {};
        v8f acc1 = {};
#pragma unroll
        for (int s = 0; s < 32; s += 2) {
            v2f a0 = *(const v2f*)(ctxRow + 4 * s + 2 * half);
            v2f a1 = *(const v2f*)(ctxRow + 4 * (s + 1) + 2 * half);
            acc0 = __builtin_amdgcn_wmma_f32_16x16x4_f32(
                false, a0, false, bfrag[s],     (short)0, acc0, false, false);
            acc1 = __builtin_amdgcn_wmma_f32_16x16x4_f32(
                false, a1, false, bfrag[s + 1], (short)0, acc1, false, false);
        }

        // Pull this lane's diagonal score out of the accumulator pair.
        float sv = 0.0f;
#pragma unroll
        for (int i = 0; i < 8; ++i) {
            float d = acc0[i] + acc1[i];
            sv = (r == i) ? d : sv;
        }

        int  jd    = t * 16 + m;
        bool valid = active && (jd < NCTX);
        float x    = (jd < C_CTX) ? sv : -sv;   // negatives use -score
        lsum += valid ? log_sigmoid(x) : 0.0f;
    }

    // wave32 reduction (deterministic)
#pragma unroll
    for (int off = 16; off > 0; off >>= 1)
        lsum += __shfl_xor(lsum, off, 32);
    if (lane == 0) wave_sum[wave] = lsum;
    __syncthreads();
    if (threadIdx.x == 0) {
        float s = 0.0f;
#pragma unroll
        for (int w = 0; w < WPB; ++w) s += wave_sum[w];
        partial[blockIdx.x] = s;
    }
}

__global__ __launch_bounds__(NBLK) void sgns_reduce_kernel(
    const float* __restrict__ partial, float* __restrict__ out)
{
    __shared__ float sm[NBLK];
    sm[threadIdx.x] = partial[threadIdx.x];
    __syncthreads();
#pragma unroll
    for (int off = NBLK / 2; off > 0; off >>= 1) {
        if (threadIdx.x < off) sm[threadIdx.x] += sm[threadIdx.x + off];
        __syncthreads();
    }
    if (threadIdx.x == 0)
        out[0] = -sm[0] / (float)(C_CTX * B_SZ);
}

extern "C" void kernel_launch(void* const* d_in, const int* in_sizes, int n_in,
                              void* d_out, int out_size, void* d_ws, size_t ws_size,
                              hipStream_t stream) {
    const float* emb_i  = (const float*)d_in[0];
    const float* emb_o  = (const float*)d_in[1];
    const int*   iword  = (const int*)d_in[2];
    const int*   owords = (const int*)d_in[3];
    const int*   nwords = (const int*)d_in[4];
    float* out     = (float*)d_out;
    float* partial = (float*)d_ws;   // NBLK floats of scratch

    sgns_score_kernel<<<NBLK, 256, 0, stream>>>(emb_i, emb_o, iword, owords,
                                                nwords, partial);
    sgns_reduce_kernel<<<1, NBLK, 0, stream>>>(partial, out);
}